// Pendulum2_L_dae_20418274525412
// MI455X (gfx1250) — compile-verified
//
#include <hip/hip_runtime.h>
#include <hip/hip_bf16.h>

// Double-pendulum constrained-dynamics RHS, closed form.
// Memory-bound streaming kernel: 32B in / 32B out per sample, ~45 flops.
// Layout per sample i (row-major (bs,8)): coords[8i..8i+3]=x, [8i+4..8i+7]=v.
// Output per sample: out[8i..8i+3]=v, out[8i+4..8i+7]=a.

__global__ __launch_bounds__(256) void pendulum2_dae_kernel(
    const float* __restrict__ coords, float* __restrict__ out, int n) {
  int i = blockIdx.x * blockDim.x + threadIdx.x;
  if (i >= n) return;

  const float4* __restrict__ cp = (const float4*)coords;
  float4* __restrict__ op = (float4*)out;

  // Prefetch ~1KB ahead on the streaming input (gfx1250 global_prefetch_b8).
  __builtin_prefetch(cp + 2 * i + 64, 0, 1);

  float4 x = cp[2 * i];       // x0 x1 x2 x3  (global_load_b128)
  float4 v = cp[2 * i + 1];   // v0 v1 v2 v3  (global_load_b128)

  float dx = x.x - x.z;
  float dy = x.y - x.w;
  float du = v.x - v.z;
  float dv = v.y - v.w;

  // L = 0.1 * phi_q * phi_q^T   (2x2, symmetric)
  float L11 = 0.4f * (x.x * x.x + x.y * x.y);
  float L12 = 0.4f * (x.x * dx + x.y * dy);
  float L22 = 0.8f * (dx * dx + dy * dy);

  // R = 0.1*phi_q*F + phi_qq*v ; phi_q row2 . F cancels exactly.
  float R1 = -20.0f * x.y + 2.0f * (v.x * v.x + v.y * v.y);
  float R2 = 2.0f * (du * du + dv * dv);

  // Solve L * lam = R (Cramer, 2x2)
  float det = L11 * L22 - L12 * L12;
  float inv = 1.0f / det;
  float lam1 = (L22 * R1 - L12 * R2) * inv;
  float lam2 = (L11 * R2 - L12 * R1) * inv;

  // a = Minv * (F - phi_q^T lam) = 0.1 * a_R
  float4 a;
  a.x = 0.1f * (-(2.0f * x.x * lam1 + 2.0f * dx * lam2));
  a.y = 0.1f * (-100.0f - (2.0f * x.y * lam1 + 2.0f * dy * lam2));
  a.z = 0.1f * (2.0f * dx * lam2);
  a.w = 0.1f * (-100.0f + 2.0f * dy * lam2);

  op[2 * i]     = v;   // global_store_b128
  op[2 * i + 1] = a;   // global_store_b128
}

extern "C" void kernel_launch(void* const* d_in, const int* in_sizes, int n_in,
                              void* d_out, int out_size, void* d_ws, size_t ws_size,
                              hipStream_t stream) {
  // d_in[0] = t (unused, size 1), d_in[1] = coords (bs*8 floats)
  const float* coords = (const float*)d_in[1];
  float* out = (float*)d_out;
  int n = in_sizes[1] / 8;  // batch size (2,097,152)

  int block = 256;
  int grid = (n + block - 1) / block;
  pendulum2_dae_kernel<<<grid, block, 0, stream>>>(coords, out, n);
}